// TreeLRU_86191403696650
// MI455X (gfx1250) — compile-verified
//
#include <hip/hip_runtime.h>

#define IN_F    128
#define DEPTH   14
#define N_NODES 16383              // 2^14 - 1
#define BATCH   8
#define ROWS    (BATCH * N_NODES)  // 131064
#define KDIM    128
#define NCOL    128
#define LDS_PITCH 132              // 132 floats = 528 B = 33*16 -> b128 aligned, bank-spread

typedef __attribute__((ext_vector_type(2))) float v2f;
typedef __attribute__((ext_vector_type(8))) float v8f;

// ---------------------------------------------------------------------------
// Preorder rank of heap-indexed node n in a complete binary tree of DEPTH.
// Walk path bits: left child -> +1, right child -> +1 + (left subtree size)
//               = +2^(DEPTH-1-t) at step t.
// ---------------------------------------------------------------------------
__device__ __forceinline__ int preorder_pos(int n) {
  unsigned b = (unsigned)(n + 1);
  int d = 31 - __clz(b);           // depth of node (root = 0)
  int pos = 0;
  for (int j = d - 1; j >= 0; --j) {
    int t = d - 1 - j;             // step index from root
    if ((b >> j) & 1u) pos += 1 << (DEPTH - 1 - t);
    else               pos += 1;
  }
  return pos;
}

// ---------------------------------------------------------------------------
// Prep: Mcat = [gamma*(B_re@Wp) ; gamma*(B_im@Wp)]  (128x128, row n = out col)
//       C2   = [C_re | -C_im]                        (128x128)
//       ubias[128], Lam[128] = {Lam_re[64], Lam_im[64]}
// grid 64x256 = 16384 threads, one weight element each.
// ---------------------------------------------------------------------------
__global__ void prep_kernel(const float* __restrict__ Wp,  const float* __restrict__ bp,
                            const float* __restrict__ nu_log, const float* __restrict__ theta_log,
                            const float* __restrict__ gamma_log,
                            const float* __restrict__ B_re, const float* __restrict__ B_im,
                            const float* __restrict__ C_re, const float* __restrict__ C_im,
                            float* __restrict__ Mcat, float* __restrict__ C2,
                            float* __restrict__ ubias, float* __restrict__ Lam)
{
  int tid = blockIdx.x * blockDim.x + threadIdx.x;   // 0..16383
  int s = tid >> 7;                                  // 0..127 (output column)
  int t = tid & 127;                                 // 0..127 (k index)
  int sr = s & 63;
  const float* Brow = ((s < 64) ? B_re : B_im) + sr * IN_F;
  float g = expf(gamma_log[sr]);
  float acc = 0.f;
  #pragma unroll 4
  for (int f = 0; f < IN_F; ++f) acc += Brow[f] * Wp[f * IN_F + t];
  Mcat[tid] = g * acc;

  C2[tid] = (t < 64) ? C_re[s * 64 + t] : -C_im[s * 64 + (t - 64)];

  if (tid < 128) {
    int ss = tid & 63;
    const float* Br2 = ((tid < 64) ? B_re : B_im) + ss * IN_F;
    float g2 = expf(gamma_log[ss]);
    float bacc = 0.f;
    for (int f = 0; f < IN_F; ++f) bacc += Br2[f] * bp[f];
    ubias[tid] = g2 * bacc;
  }
  if (tid < 64) {
    float lm = expf(-expf(nu_log[tid]));
    float th = expf(theta_log[tid]);
    Lam[tid]      = lm * cosf(th);
    Lam[64 + tid] = lm * sinf(th);
  }
}

// ---------------------------------------------------------------------------
// Row-GEMM: out[r, :] = in[r, :] @ W.T (+ bias), W stored as [n][k] 128x128.
// Block = 8 waves, each wave owns a 16-row strip x 128 cols.
// V_WMMA_F32_16X16X4_F32, K = 32 steps of 4.
// SCATTER: out row = b*N_NODES + preorder_pos(n)  (output stage)
// ---------------------------------------------------------------------------
template <bool SCATTER, bool HAS_BIAS>
__global__ __launch_bounds__(256)
void gemm_kernel(const float* __restrict__ in, const float* __restrict__ W,
                 const float* __restrict__ bias, float* __restrict__ out)
{
  __shared__ float ldsW[NCOL * LDS_PITCH];   // 67,584 B

  // Cooperative staged copy of W (row-major 128x128) into pitched LDS, 16B chunks.
  {
    const float4* Wv = (const float4*)W;
    for (int i = threadIdx.x; i < NCOL * (KDIM / 4); i += 256) {
      int n  = i >> 5;          // /32 float4 per row
      int k4 = i & 31;
      *(float4*)&ldsW[n * LDS_PITCH + k4 * 4] = Wv[i];
    }
    __syncthreads();
  }

  const int lane  = threadIdx.x & 31;
  const int wave  = threadIdx.x >> 5;
  const int rowTile = blockIdx.x * 128 + wave * 16;

  const int m     = lane & 15;          // M (A) / N (B,C) index within tile
  const int khalf = (lane >> 4) << 1;   // 0 or 2 : K sub-offset per half-wave

  int arow = rowTile + m;
  if (arow >= ROWS) arow = ROWS - 1;    // clamp (EXEC must stay all-ones for WMMA)
  const float* aptr = in + (size_t)arow * KDIM + khalf;

  v8f acc[8];
  #pragma unroll
  for (int ct = 0; ct < 8; ++ct) {
    float bv = HAS_BIAS ? bias[ct * 16 + m] : 0.f;
    #pragma unroll
    for (int v = 0; v < 8; ++v) acc[ct][v] = bv;
  }

  #pragma unroll 2
  for (int kk = 0; kk < 32; ++kk) {
    const int kb = kk * 4;
    v2f a = *(const v2f*)(aptr + kb);                       // A: M=m, K=kb+khalf{+0,+1}
    #pragma unroll
    for (int ct = 0; ct < 8; ++ct) {
      v2f b = *(const v2f*)&ldsW[(ct * 16 + m) * LDS_PITCH + kb + khalf];
      acc[ct] = __builtin_amdgcn_wmma_f32_16x16x4_f32(
          false, a, false, b, (short)0, acc[ct], false, false);
    }
  }

  // Epilogue: C layout -> VGPR v holds row (v + 8*(lane/16)), col = ct*16 + m.
  #pragma unroll
  for (int v = 0; v < 8; ++v) {
    int gr = rowTile + v + ((lane >> 4) << 3);
    if (gr < ROWS) {
      int orow;
      if (SCATTER) {
        int b = gr / N_NODES;
        int n = gr - b * N_NODES;
        orow = b * N_NODES + preorder_pos(n);
      } else {
        orow = gr;
      }
      float* op = out + (size_t)orow * NCOL;
      #pragma unroll
      for (int ct = 0; ct < 8; ++ct) op[ct * 16 + m] = acc[ct][v];
    }
  }
}

// ---------------------------------------------------------------------------
// Tree scan, one level (in place on U; U row layout: [re[64] | im[64]]).
// h[node] = Lam*(h[2n+1] + h[2n+2]) + u[node]
// ---------------------------------------------------------------------------
__device__ __forceinline__ void scan_update(float* __restrict__ U,
                                            const float* __restrict__ Lam,
                                            size_t base, int node, int s)
{
  float* selfp      = U + (base + node) * 128;
  const float* lp   = U + (base + 2 * node + 1) * 128;
  const float* rp   = U + (base + 2 * node + 2) * 128;
  float sre = lp[s]      + rp[s];
  float sim = lp[64 + s] + rp[64 + s];
  float Lre = Lam[s], Lim = Lam[64 + s];
  selfp[s]      += Lre * sre - Lim * sim;
  selfp[64 + s] += Lre * sim + Lim * sre;
}

__global__ void scan_level_kernel(float* __restrict__ U, const float* __restrict__ Lam, int d)
{
  int tid = blockIdx.x * blockDim.x + threadIdx.x;
  int cnt = 1 << d;
  if (tid >= ((BATCH * 64) << d)) return;
  int s = tid & 63;
  int j = (tid >> 6) & (cnt - 1);
  int b = tid >> (6 + d);
  scan_update(U, Lam, (size_t)b * N_NODES, (cnt - 1) + j, s);
}

// Levels 6..0 fused: one block per batch, barrier between levels.
__global__ void scan_top_kernel(float* __restrict__ U, const float* __restrict__ Lam)
{
  size_t base = (size_t)blockIdx.x * N_NODES;
  for (int d = 6; d >= 0; --d) {
    int cnt = 1 << d;
    int total = cnt << 6;
    for (int e = threadIdx.x; e < total; e += blockDim.x) {
      int s = e & 63;
      int j = e >> 6;
      scan_update(U, Lam, base, (cnt - 1) + j, s);
    }
    __syncthreads();
  }
}

// ---------------------------------------------------------------------------
extern "C" void kernel_launch(void* const* d_in, const int* in_sizes, int n_in,
                              void* d_out, int out_size, void* d_ws, size_t ws_size,
                              hipStream_t stream)
{
  const float* x         = (const float*)d_in[0];
  const float* Wp        = (const float*)d_in[1];
  const float* bp        = (const float*)d_in[2];
  const float* nu_log    = (const float*)d_in[3];
  const float* theta_log = (const float*)d_in[4];
  const float* gamma_log = (const float*)d_in[5];
  const float* B_re      = (const float*)d_in[6];
  const float* B_im      = (const float*)d_in[7];
  const float* C_re      = (const float*)d_in[8];
  const float* C_im      = (const float*)d_in[9];
  float* out = (float*)d_out;

  char*  ws    = (char*)d_ws;
  float* Mcat  = (float*)ws;                 // 16384 f
  float* C2    = Mcat + 16384;               // 16384 f
  float* ubias = C2 + 16384;                 // 128 f
  float* Lam   = ubias + 128;                // 128 f
  float* U     = (float*)(ws + 256 * 1024);  // ROWS*128 f (~64 MB)

  prep_kernel<<<64, 256, 0, stream>>>(Wp, bp, nu_log, theta_log, gamma_log,
                                      B_re, B_im, C_re, C_im, Mcat, C2, ubias, Lam);

  // u = x @ Mcat.T + ubias   (131064 x 128 x 128)
  gemm_kernel<false, true><<<1024, 256, 0, stream>>>(x, Mcat, ubias, U);

  // bottom-up tree scan (leaves at d=13 are u itself)
  for (int d = 12; d >= 7; --d) {
    int total = (BATCH * 64) << d;
    scan_level_kernel<<<(total + 255) / 256, 256, 0, stream>>>(U, Lam, d);
  }
  scan_top_kernel<<<BATCH, 256, 0, stream>>>(U, Lam);

  // y = [h_re|h_im] @ [C_re|-C_im].T, scattered to preorder positions
  gemm_kernel<true, false><<<1024, 256, 0, stream>>>(U, C2, nullptr, out);
}